// DualGCN_49143015801442
// MI455X (gfx1250) — compile-verified
//
#include <hip/hip_runtime.h>
#include <hip/hip_bf16.h>

// ---------------------------------------------------------------------------
// DualGCN for MI455X (gfx1250, wave32).
// Dense layers via V_WMMA_F32_16X16X4_F32 (fp32 precision, matches reference).
// SpMM via CSR-by-row waves (deterministic, no atomics). Concats eliminated by
// writing directly into strided slots of wide workspace matrices.
// ---------------------------------------------------------------------------

typedef __attribute__((ext_vector_type(2))) float v2f;
typedef __attribute__((ext_vector_type(8))) float v8f;

#define LDW 132   // LDS row stride in floats: bank stride 4 -> conflict-free b64 reads

// ---- CSR row pointers from sorted COO rows: rp[t] = lower_bound(row, t) ----
__global__ __launch_bounds__(256) void rowptr_kernel(const int* __restrict__ row,
                                                     int* __restrict__ rp,
                                                     int n, int e) {
    int t = blockIdx.x * blockDim.x + threadIdx.x;
    if (t > n) return;
    int lo = 0, hi = e;
    while (lo < hi) {
        int mid = (lo + hi) >> 1;
        if (row[mid] < t) lo = mid + 1; else hi = mid;
    }
    rp[t] = lo;
}

// ---- SpMM: Y[r, cofs:cofs+128] = sum_e val[e] * X[col[e], 0:128], one wave/row ----
__global__ __launch_bounds__(256) void spmm_kernel(const int* __restrict__ rp,
                                                   const int* __restrict__ col,
                                                   const float* __restrict__ val,
                                                   const float* __restrict__ X, int ldx,
                                                   float* __restrict__ Y, int ldy, int cofs,
                                                   int n) {
    int w    = (blockIdx.x * blockDim.x + threadIdx.x) >> 5;   // wave id == row id
    int lane = threadIdx.x & 31;
    if (w >= n) return;
    int e0 = rp[w], e1 = rp[w + 1];
    float4 acc = make_float4(0.f, 0.f, 0.f, 0.f);
    for (int e = e0; e < e1; e += 32) {
        int cnt = e1 - e; if (cnt > 32) cnt = 32;
        int   c = 0;  float v = 0.f;
        if (lane < cnt) { c = col[e + lane]; v = val[e + lane]; }
        for (int j = 0; j < cnt; ++j) {
            int   cj = __shfl(c, j, 32);
            float vj = __shfl(v, j, 32);
            const float4* xr = (const float4*)(X + (size_t)cj * ldx);
            float4 xv = xr[lane];                       // coalesced 512B row gather (L2-resident)
            acc.x = fmaf(vj, xv.x, acc.x);
            acc.y = fmaf(vj, xv.y, acc.y);
            acc.z = fmaf(vj, xv.z, acc.z);
            acc.w = fmaf(vj, xv.w, acc.w);
        }
    }
    float4* yr = (float4*)(Y + (size_t)w * ldy + cofs);
    yr[lane] = acc;
}

// ---- GEMM: Y[m, cofs+n] = relu?(X[m,:] . W[n,:] + bias[n]) via WMMA f32 16x16x4 ----
// Block: 256 thr = 8 waves; each wave owns a 16-row strip and NTILES 16-col tiles.
// Weights staged in LDS in 128-deep K chunks. NTILES is compile-time so the
// inner loop is branch-free: batch all B ds_loads, one wait, NTILES WMMAs.
template <int NTILES>
__global__ __launch_bounds__(256) void gemm_kernel(const float* __restrict__ X, int ldx,
                                                   const float* __restrict__ W,     // [NTILES*16, K]
                                                   const float* __restrict__ bias,  // [NTILES*16]
                                                   float* __restrict__ Y, int ldy, int cofs,
                                                   int M, int K, int do_relu) {
    constexpr int NOUT = NTILES * 16;
    __shared__ float Wl[NOUT * LDW];

    const int tid  = threadIdx.x;
    const int wid  = tid >> 5;
    const int lane = tid & 31;
    const int half = lane >> 4;       // lane half selects K (A/B) or M (C/D) split
    const int l    = lane & 15;
    const int m0   = blockIdx.x * 128 + wid * 16;
    int mrow = m0 + l; if (mrow >= M) mrow = M - 1;   // clamp: keep EXEC all-1s for WMMA

    v8f acc[NTILES] = {};

    for (int k0 = 0; k0 < K; k0 += 128) {
        __syncthreads();
        // cooperative coalesced load of W[:, k0:k0+128) into padded LDS
        for (int idx = tid; idx < NOUT * 128; idx += 256) {
            int n = idx >> 7, kk = idx & 127;
            Wl[n * LDW + kk] = W[(size_t)n * K + k0 + kk];
        }
        __syncthreads();

        const float* xrow  = X + (size_t)mrow * ldx + k0 + 2 * half;
        const float* wbase = &Wl[l * LDW + 2 * half];

#pragma unroll 4
        for (int kk = 0; kk < 128; kk += 4) {
            // A 16x4 fp32: v0 = K{0|2}, v1 = K{1|3} split by lane half
            v2f a;
            {
                const float2 t2 = *(const float2*)(xrow + kk);
                a[0] = t2.x; a[1] = t2.y;
            }
            // batch all B fragment loads first (one dscnt wait, then WMMA burst)
            v2f b[NTILES];
#pragma unroll
            for (int j = 0; j < NTILES; ++j) {
                const float2 w2 = *(const float2*)(wbase + j * 16 * LDW + kk);
                b[j][0] = w2.x; b[j][1] = w2.y;
            }
#pragma unroll
            for (int j = 0; j < NTILES; ++j) {
                acc[j] = __builtin_amdgcn_wmma_f32_16x16x4_f32(
                    /*neg_a=*/false, a, /*neg_b=*/false, b[j],
                    /*c_mod=*/(short)0, acc[j], /*reuse_a=*/false, /*reuse_b=*/false);
            }
        }
    }

    // epilogue: C/D layout -> element i of acc is row (m0 + i + 8*half), col (tile*16 + l)
#pragma unroll
    for (int j = 0; j < NTILES; ++j) {
        const float bj = bias[j * 16 + l];
#pragma unroll
        for (int i = 0; i < 8; ++i) {
            int m = m0 + i + 8 * half;
            if (m < M) {
                float vv = acc[j][i] + bj;
                if (do_relu) vv = fmaxf(vv, 0.0f);
                Y[(size_t)m * ldy + cofs + j * 16 + l] = vv;
            }
        }
    }
}

extern "C" void kernel_launch(void* const* d_in, const int* in_sizes, int n_in,
                              void* d_out, int out_size, void* d_ws, size_t ws_size,
                              hipStream_t stream) {
    const float* x    = (const float*)d_in[0];
    const int*   rowa = (const int*)  d_in[1];
    const int*   cola = (const int*)  d_in[2];
    const float* vala = (const float*)d_in[3];
    const int*   rowb = (const int*)  d_in[4];
    const int*   colb = (const int*)  d_in[5];
    const float* valb = (const float*)d_in[6];
    const float* Wa0  = (const float*)d_in[7];   const float* ba0 = (const float*)d_in[8];
    const float* Wa1  = (const float*)d_in[9];   const float* ba1 = (const float*)d_in[10];
    const float* Wb0  = (const float*)d_in[11];  const float* bb0 = (const float*)d_in[12];
    const float* Wb1  = (const float*)d_in[13];  const float* bb1 = (const float*)d_in[14];
    const float* Wm   = (const float*)d_in[15];  const float* bm  = (const float*)d_in[16];
    const float* Wo   = (const float*)d_in[17];  const float* bo  = (const float*)d_in[18];
    float* out = (float*)d_out;

    const int N = in_sizes[0] / 128;
    const int E = in_sizes[1];

    // workspace carve-up (~180 MB): H0[N,128] S[N,128] G[N,384] C[N,256] rpA/rpB[N+1]
    float* H0 = (float*)d_ws;
    float* S  = H0 + (size_t)N * 128;
    float* G  = S  + (size_t)N * 128;      // g0 | g1 | g2 written in place (no concat copy)
    float* C  = G  + (size_t)N * 384;      // x_a | x_b written in place
    int*   rpA = (int*)(C + (size_t)N * 256);
    int*   rpB = rpA + (N + 1);

    const int gRp = (N + 256) / 256;       // N+1 threads
    const int gM  = (N + 127) / 128;       // 128-row strips
    const int gSp = (N + 7) / 8;           // 8 waves (rows) per block

    rowptr_kernel<<<gRp, 256, 0, stream>>>(rowa, rpA, N, E);
    rowptr_kernel<<<gRp, 256, 0, stream>>>(rowb, rpB, N, E);

    // ---- branch a ----
    gemm_kernel<8><<<gM, 256, 0, stream>>>(x, 128, Wa0, ba0, H0, 128, 0, N, 128, 1);  // h0
    spmm_kernel<<<gSp, 256, 0, stream>>>(rpA, cola, vala, H0, 128, S, 128, 0, N);     // A@h0
    gemm_kernel<8><<<gM, 256, 0, stream>>>(S, 128, Wa1, ba1, H0, 128, 0, N, 128, 1);  // h1
    spmm_kernel<<<gSp, 256, 0, stream>>>(rpA, cola, vala, H0, 128, C, 256, 0, N);     // x_a -> C[:,0:128]

    // ---- branch b ----
    gemm_kernel<8><<<gM, 256, 0, stream>>>(x, 128, Wb0, bb0, G, 384, 0, N, 128, 1);     // g0 -> G[:,0:128]
    spmm_kernel<<<gSp, 256, 0, stream>>>(rpB, colb, valb, G, 384, S, 128, 0, N);        // B@g0
    gemm_kernel<8><<<gM, 256, 0, stream>>>(S, 128, Wb1, bb1, G, 384, 128, N, 128, 1);   // g1 -> G[:,128:256]
    spmm_kernel<<<gSp, 256, 0, stream>>>(rpB, colb, valb, G + 128, 384, G, 384, 256, N);// g2 -> G[:,256:384]
    gemm_kernel<8><<<gM, 256, 0, stream>>>(G, 384, Wm, bm, C, 256, 128, N, 384, 1);     // x_b -> C[:,128:256]

    // ---- merge ----
    gemm_kernel<4><<<gM, 256, 0, stream>>>(C, 256, Wo, bo, out, 64, 0, N, 256, 0);      // out [N,64]
}